// QuantumLayer_65481071399210
// MI455X (gfx1250) — compile-verified
//
#include <hip/hip_runtime.h>
#include <stdint.h>

// Problem constants (match reference: B=262144, Q=32, L=8)
#define QB_B 262144
#define QB_Q 32
#define QB_L 8

// ---------------------------------------------------------------------------
// Kernel 1: collapse the 24 per-qubit parameter gates into one 2x2 complex
// unitary M_q (fp64 for accuracy), then reduce to the two closed-form
// coefficients:  out = A*cos(x) + C*sin(x).
// Track the two COLUMNS of M: col0 = (m00,m10), col1 = (m01,m11); applying a
// gate to M on the left is applying it to each column as a 2-amp state.
// ---------------------------------------------------------------------------
__global__ void qcoef_kernel(const float* __restrict__ params,
                             float2* __restrict__ ac) {
  int q = threadIdx.x;
  if (q >= QB_Q) return;
  // col0 = (1,0), col1 = (0,1)  (complex: r/i parts)
  double a0r = 1.0, a0i = 0.0, b0r = 0.0, b0i = 0.0;
  double a1r = 0.0, a1i = 0.0, b1r = 1.0, b1i = 0.0;
  for (int l = 0; l < QB_L; ++l) {
    double t0 = (double)params[(l * QB_Q + q) * 3 + 0];
    double t1 = (double)params[(l * QB_Q + q) * 3 + 1];
    double t2 = (double)params[(l * QB_Q + q) * 3 + 2];
    double c, s, ar, ai, br, bi;
    // RY(t0): a' = c*a - s*b ; b' = s*a + c*b   (real matrix)
    c = cos(t0 * 0.5); s = sin(t0 * 0.5);
    ar = a0r; ai = a0i; br = b0r; bi = b0i;
    a0r = c*ar - s*br; a0i = c*ai - s*bi; b0r = s*ar + c*br; b0i = s*ai + c*bi;
    ar = a1r; ai = a1i; br = b1r; bi = b1i;
    a1r = c*ar - s*br; a1i = c*ai - s*bi; b1r = s*ar + c*br; b1i = s*ai + c*bi;
    // RZ(t1): a' = e^{-it/2} a ; b' = e^{+it/2} b
    c = cos(t1 * 0.5); s = sin(t1 * 0.5);
    ar = a0r; ai = a0i; br = b0r; bi = b0i;
    a0r = c*ar + s*ai; a0i = c*ai - s*ar; b0r = c*br - s*bi; b0i = c*bi + s*br;
    ar = a1r; ai = a1i; br = b1r; bi = b1i;
    a1r = c*ar + s*ai; a1i = c*ai - s*ar; b1r = c*br - s*bi; b1i = c*bi + s*br;
    // RX(t2): a' = c*a - i s*b ; b' = -i s*a + c*b
    c = cos(t2 * 0.5); s = sin(t2 * 0.5);
    ar = a0r; ai = a0i; br = b0r; bi = b0i;
    a0r = c*ar + s*bi; a0i = c*ai - s*br; b0r = s*ai + c*br; b0i = -s*ar + c*bi;
    ar = a1r; ai = a1i; br = b1r; bi = b1i;
    a1r = c*ar + s*bi; a1i = c*ai - s*br; b1r = s*ai + c*br; b1i = -s*ar + c*bi;
  }
  // A = |m00|^2 - |m10|^2 ; C = Im(m10*conj(m11)) - Im(m00*conj(m01))
  double A = a0r * a0r + a0i * a0i - b0r * b0r - b0i * b0i;
  double C = (b0i * b1r - b0r * b1i) - (a0i * a1r - a0r * a1i);
  ac[q] = make_float2((float)A, (float)C);
}

// ---------------------------------------------------------------------------
// Kernel 2: streaming evaluation  out = A_q*cos(x) + C_q*sin(x).
// HBM-bound (64 MB total traffic). Uses gfx1250 async global->LDS loads with
// ASYNCcnt double buffering for the x stream.
// Exact coverage: 2048 blocks * 256 thr * 4 iters * float4 = 2^23 elements.
// ---------------------------------------------------------------------------
#define QZ_BLOCK 256
#define QZ_ITERS 4

__global__ __launch_bounds__(QZ_BLOCK) void qz_kernel(
    const float* __restrict__ x, const float2* __restrict__ ac,
    float* __restrict__ out) {
  __shared__ float4 tile[2][QZ_BLOCK];
  const unsigned tid = threadIdx.x;
  const unsigned f4base = blockIdx.x * QZ_BLOCK + tid;   // float4 index
  const unsigned strideF4 = gridDim.x * QZ_BLOCK;        // float4 per iter

  // Per-thread qubit indices are fixed: element idx = 4*f4, qubit = idx % 32,
  // and stride (2^21 elements) is a multiple of 32.
  const int qb = (int)(tid & 7u) * 4;
  const float A0 = ac[qb + 0].x, C0 = ac[qb + 0].y;
  const float A1 = ac[qb + 1].x, C1 = ac[qb + 1].y;
  const float A2 = ac[qb + 2].x, C2 = ac[qb + 2].y;
  const float A3 = ac[qb + 3].x, C3 = ac[qb + 3].y;

  const unsigned long long sbase = (unsigned long long)x;
  // LDS byte offsets (low 32 bits of the generic pointer = wave-relative LDS
  // address, which is what the async-load VDST expects).
  const unsigned lds0 = (unsigned)(size_t)&tile[0][tid];
  const unsigned lds1 = (unsigned)(size_t)&tile[1][tid];

  // Prime the pipeline: async load of tile 0 into buffer 0.
  {
    unsigned voff = f4base * 16u;
    asm volatile("global_load_async_to_lds_b128 %0, %1, %2"
                 :: "v"(lds0), "v"(voff), "s"(sbase) : "memory");
  }

  float4* out4 = (float4*)out;
#pragma unroll
  for (int it = 0; it < QZ_ITERS; ++it) {
    if (it + 1 < QZ_ITERS) {
      // WAR safety: make sure our earlier ds_load of this buffer has
      // completed before the async engine may overwrite it.
      asm volatile("s_wait_dscnt 0x0" ::: "memory");
      unsigned voff = (f4base + (unsigned)(it + 1) * strideF4) * 16u;
      unsigned ldsn = ((it + 1) & 1) ? lds1 : lds0;
      asm volatile("global_load_async_to_lds_b128 %0, %1, %2"
                   :: "v"(ldsn), "v"(voff), "s"(sbase) : "memory");
      // Async loads complete in order: <=1 outstanding => current tile ready.
      asm volatile("s_wait_asynccnt 0x1" ::: "memory");
    } else {
      asm volatile("s_wait_asynccnt 0x0" ::: "memory");
    }
    // Each lane reads back exactly the 16 bytes it async-loaded: no barrier.
    float4 v = tile[it & 1][tid];
    float s, c;
    float4 r;
    __sincosf(v.x, &s, &c); r.x = fmaf(A0, c, C0 * s);
    __sincosf(v.y, &s, &c); r.y = fmaf(A1, c, C1 * s);
    __sincosf(v.z, &s, &c); r.z = fmaf(A2, c, C2 * s);
    __sincosf(v.w, &s, &c); r.w = fmaf(A3, c, C3 * s);
    out4[f4base + (unsigned)it * strideF4] = r;
  }
}

extern "C" void kernel_launch(void* const* d_in, const int* in_sizes, int n_in,
                              void* d_out, int out_size, void* d_ws,
                              size_t ws_size, hipStream_t stream) {
  (void)in_sizes; (void)n_in; (void)out_size; (void)ws_size;
  const float* x = (const float*)d_in[0];       // (B, Q) float32
  const float* params = (const float*)d_in[1];  // (L, Q, 3) float32
  float* out = (float*)d_out;                   // (B, Q) float32
  float2* ac = (float2*)d_ws;                   // 32 coefficient pairs (256 B)

  qcoef_kernel<<<1, 32, 0, stream>>>(params, ac);

  // 2048 blocks * 256 threads * 4 float4-iters = 2^21 float4 = B*Q elements.
  qz_kernel<<<2048, QZ_BLOCK, 0, stream>>>(x, ac, out);
}